// EncoderLayer_19189913878730
// MI455X (gfx1250) — compile-verified
//
#include <hip/hip_runtime.h>

// GRU (reset_after=True): B=32, T=512, D=512, H=1024
// Phase 1: xw = x @ W + b_i           (big parallel GEMM, fp32 WMMA)
// Phase 2: 512 sequential step kernels (h @ U from L2-resident U, fp32 WMMA)

#define B_  32
#define T_  512
#define D_  512
#define H_  1024
#define H3_ 3072

typedef __attribute__((ext_vector_type(2))) float v2f;
typedef __attribute__((ext_vector_type(8))) float v8f;

// ---------------------------------------------------------------------------
// Phase 1: xw[BT,3H] = x[BT,D] @ W[D,3H] + b_i
// grid = (BT/16, 3H/128), block = 256 (8 waves); wave w owns N-subtile w.
// A(16x4 f32): lane(l<16) holds row=l, K={k,k+1}; lane(l>=16) row=l-16, K={k+2,k+3}
// B(4x16 f32): lane(l<16) holds col=l, K={k,k+1}; lane(l>=16) col=l-16, K={k+2,k+3}
// C/D(16x16 f32): vgpr v, lane l -> (M = v + 8*(l/16), N = l%16)
// ---------------------------------------------------------------------------
__global__ __launch_bounds__(256)
void gru_xw_gemm(const float* __restrict__ x, const float* __restrict__ W,
                 const float* __restrict__ b, float* __restrict__ xw) {
  const int wave = threadIdx.x >> 5;
  const int lane = threadIdx.x & 31;
  const int half = lane >> 4;
  const int lr   = lane & 15;
  const int m0 = blockIdx.x * 16;
  const int n0 = blockIdx.y * 128 + wave * 16;

  const float* Arow = x + (size_t)(m0 + lr) * D_;   // this lane's A row
  const float* Bcol = W + n0 + lr;                  // this lane's B column
  v8f c = {};
  #pragma unroll 4
  for (int k = 0; k < D_; k += 4) {
    const int ka = k + 2 * half;
    v2f a;  a.x  = Arow[ka];
            a.y  = Arow[ka + 1];
    v2f bb; bb.x = Bcol[(size_t)ka * H3_];
            bb.y = Bcol[(size_t)(ka + 1) * H3_];
    c = __builtin_amdgcn_wmma_f32_16x16x4_f32(false, a, false, bb,
                                              (short)0, c, false, false);
  }
  const float bias = b[n0 + lr];                    // b_i (row 0 of b)
  float* out = xw + (size_t)m0 * H3_ + n0 + lr;
  #pragma unroll
  for (int v = 0; v < 8; ++v) {
    const int m = v + 8 * half;
    out[(size_t)m * H3_] = c[v] + bias;
  }
}

// ---------------------------------------------------------------------------
// Phase 2: one recurrence step.
// grid = H/16 = 64 blocks; block = 384 threads = 12 waves.
// Wave role: tile = wave%6 -> (mhalf = tile&1, gate = tile>>1), khalf = wave/6.
// Each wave: 128 fp32 WMMAs over its 512-wide K range; K-halves reduced in LDS.
// Then all threads compute the GRU gates for this block's 32x16 h tile.
// ---------------------------------------------------------------------------
__global__ __launch_bounds__(384)
void gru_step(const float* __restrict__ U, const float* __restrict__ b,
              const float* __restrict__ xw, const float* __restrict__ h_in,
              float* __restrict__ h_out, float* __restrict__ out, int t) {
  __shared__ float hu[6][16][16];                   // [gate*2+mhalf][m][n]
  const int wave  = threadIdx.x >> 5;
  const int lane  = threadIdx.x & 31;
  const int tile  = wave % 6;
  const int khalf = wave / 6;
  const int mh    = tile & 1;
  const int gate  = tile >> 1;
  const int half  = lane >> 4;
  const int lr    = lane & 15;
  const int n0    = blockIdx.x * 16;

  const float* Arow = h_in + (size_t)(mh * 16 + lr) * H_;
  const float* Bcol = U + (size_t)gate * H_ + n0 + lr;
  const int k0 = khalf * (H_ / 2);

  v8f c = {};
  #pragma unroll 4
  for (int k = k0; k < k0 + H_ / 2; k += 4) {
    const int ka = k + 2 * half;
    v2f a;  a.x  = Arow[ka];
            a.y  = Arow[ka + 1];
    v2f bb; bb.x = Bcol[(size_t)ka * H3_];
            bb.y = Bcol[(size_t)(ka + 1) * H3_];
    c = __builtin_amdgcn_wmma_f32_16x16x4_f32(false, a, false, bb,
                                              (short)0, c, false, false);
  }

  // reduce the two K-halves through LDS
  if (khalf == 0) {
    #pragma unroll
    for (int v = 0; v < 8; ++v) hu[tile][v + 8 * half][lr] = c[v];
  }
  __syncthreads();
  if (khalf == 1) {
    #pragma unroll
    for (int v = 0; v < 8; ++v) hu[tile][v + 8 * half][lr] += c[v];
  }
  __syncthreads();

  // gates: 32 rows x 16 cols = 512 elements
  const float* br = b + H3_;                        // b_r (row 1 of b)
  for (int i = threadIdx.x; i < 32 * 16; i += 384) {
    const int m  = i >> 4;                          // batch index
    const int n  = i & 15;
    const int m2 = m >> 4, mr = m & 15;
    const int col = n0 + n;
    const float hz = hu[0 + m2][mr][n] + br[col];
    const float hr = hu[2 + m2][mr][n] + br[H_ + col];
    const float hh = hu[4 + m2][mr][n] + br[2 * H_ + col];
    const float* xwrow = xw + ((size_t)m * T_ + t) * H3_;
    const float xz = xwrow[col];
    const float xr = xwrow[H_ + col];
    const float xh = xwrow[2 * H_ + col];
    const float z  = 1.0f / (1.0f + __expf(-(xz + hz)));
    const float r  = 1.0f / (1.0f + __expf(-(xr + hr)));
    const float hc = tanhf(xh + r * hh);
    const float hp = h_in[(size_t)m * H_ + col];
    const float hn = z * hp + (1.0f - z) * hc;
    h_out[(size_t)m * H_ + col] = hn;
    out[((size_t)m * T_ + t) * H_ + col] = hn;
  }
}

__global__ void gru_zero(float* __restrict__ p, int n) {
  const int i = blockIdx.x * blockDim.x + threadIdx.x;
  if (i < n) p[i] = 0.0f;
}

__global__ void gru_copy(const float* __restrict__ s, float* __restrict__ d, int n) {
  const int i = blockIdx.x * blockDim.x + threadIdx.x;
  if (i < n) d[i] = s[i];
}

extern "C" void kernel_launch(void* const* d_in, const int* in_sizes, int n_in,
                              void* d_out, int out_size, void* d_ws, size_t ws_size,
                              hipStream_t stream) {
  const float* x = (const float*)d_in[0];   // [B,T,D]
  const float* W = (const float*)d_in[1];   // [D,3H]
  const float* U = (const float*)d_in[2];   // [H,3H]
  const float* b = (const float*)d_in[3];   // [2,3H]

  float* out    = (float*)d_out;                       // [B,T,H] then [B,H]
  float* h_last = out + (size_t)B_ * T_ * H_;

  float* xw   = (float*)d_ws;                          // [B,T,3H] scratch
  float* hbuf = xw + (size_t)B_ * T_ * H3_;            // 2 x [B,H] double buffer

  // h0 = 0
  {
    const int n = B_ * H_;
    gru_zero<<<(n + 255) / 256, 256, 0, stream>>>(hbuf, n);
  }

  // Phase 1: xw = x@W + b_i
  {
    dim3 grid(B_ * T_ / 16, H3_ / 128);
    gru_xw_gemm<<<grid, 256, 0, stream>>>(x, W, b, xw);
  }

  // Phase 2: sequential scan; stream ordering carries the dependency.
  for (int t = 0; t < T_; ++t) {
    const float* hin  = hbuf + (size_t)(t & 1) * B_ * H_;
    float*       hout = hbuf + (size_t)((t + 1) & 1) * B_ * H_;
    gru_step<<<H_ / 16, 384, 0, stream>>>(U, b, xw, hin, hout, out, t);
  }

  // h_last lives in buffer 0 after T=512 (even) steps
  {
    const int n = B_ * H_;
    gru_copy<<<(n + 255) / 256, 256, 0, stream>>>(hbuf, h_last, n);
  }
}